// TripletLossWrapper_19731079758399
// MI455X (gfx1250) — compile-verified
//
#include <hip/hip_runtime.h>
#include <hip/hip_bf16.h>
#include <cmath>

// Triplet margin loss on MI455X (gfx1250, wave32).
//
// Per-row squared distances computed as the diagonal of a Gram matrix via
// V_WMMA_F32_16X16X4_F32:  diag(d * d^T) = row-wise sum of squares, full fp32.
// The f32 16x4 A-operand layout and the 4x16 B-operand layout of the
// transposed matrix are identical per-lane, so a single float2 register pair
// of diffs feeds both WMMA operands.

typedef __attribute__((ext_vector_type(2))) float v2f;
typedef __attribute__((ext_vector_type(8))) float v8f;

#define TL_N 4096
#define TL_D 2048
#define TL_EPS 1e-6f
#define TL_MARGIN 1.0f

__global__ void __launch_bounds__(256)
triplet_dist_kernel(const float* __restrict__ emb,
                    const int* __restrict__ pos_idx,
                    const int* __restrict__ neg_idx,
                    float* __restrict__ row_loss)
{
    const int lane = threadIdx.x & 31;      // lane in wave32
    const int wave = threadIdx.x >> 5;      // 0..7 : K-segment of this wave
    const int tile = blockIdx.x;            // 0..255 : 16-row tile
    const int m    = lane & 15;             // row within tile handled by lane
    const int h    = lane >> 4;             // 0/1 : which K-pair within chunk
    const int row  = tile * 16 + m;

    // Gathered rows (lanes 16..31 duplicate lanes 0..15 — same addresses, cached)
    const int prow = pos_idx[row];
    const int nrow = neg_idx[row];

    const float* a_ptr = emb + (size_t)row  * TL_D;
    const float* p_ptr = emb + (size_t)prow * TL_D;
    const float* n_ptr = emb + (size_t)nrow * TL_D;

    v8f acc_ap = {};   // 16x16 f32 accumulator, diagonal = ||a-p+eps||^2 per row
    v8f acc_an = {};

    const int kseg  = TL_D / 8;             // 256 columns per wave
    const int kbase = wave * kseg;
    const int kend  = kbase + kseg;

    #pragma unroll 4
    for (int k0 = kbase; k0 < kend; k0 += 4) {
        const int c = k0 + 2 * h;           // lane's two consecutive columns
        v2f a2 = *(const v2f*)(a_ptr + c);
        v2f p2 = *(const v2f*)(p_ptr + c);
        v2f n2 = *(const v2f*)(n_ptr + c);
        v2f dap = a2 - p2 + TL_EPS;
        v2f dan = a2 - n2 + TL_EPS;
        // D = A*B + C with A == B(register-wise) -> Gram accumulation
        acc_ap = __builtin_amdgcn_wmma_f32_16x16x4_f32(
            false, dap, false, dap, (short)0, acc_ap, false, false);
        acc_an = __builtin_amdgcn_wmma_f32_16x16x4_f32(
            false, dan, false, dan, (short)0, acc_an, false, false);
    }

    // Extract Gram diagonal: row m<8 lives in VGPR m @ lane m;
    // row m>=8 lives in VGPR m-8 @ lane m+16.
    float sap = 0.0f, san = 0.0f;
    const int src_byte = ((m < 8) ? m : (m + 16)) << 2;
    #pragma unroll
    for (int r = 0; r < 8; ++r) {
        float t1 = __int_as_float(
            __builtin_amdgcn_ds_bpermute(src_byte, __float_as_int(acc_ap[r])));
        float t2 = __int_as_float(
            __builtin_amdgcn_ds_bpermute(src_byte, __float_as_int(acc_an[r])));
        if ((m & 7) == r) { sap = t1; san = t2; }
    }

    // Combine the 8 K-segment partials per row in a fixed order (deterministic).
    __shared__ float s_ap[8][16];
    __shared__ float s_an[8][16];
    if (lane < 16) {
        s_ap[wave][m] = sap;
        s_an[wave][m] = san;
    }
    __syncthreads();

    if (threadIdx.x < 16) {
        float tap = 0.0f, tng = 0.0f;
        #pragma unroll
        for (int w = 0; w < 8; ++w) {
            tap += s_ap[w][threadIdx.x];
            tng += s_an[w][threadIdx.x];
        }
        float loss = sqrtf(tap) - sqrtf(tng) + TL_MARGIN;
        row_loss[tile * 16 + threadIdx.x] = fmaxf(loss, 0.0f);
    }
}

__global__ void __launch_bounds__(256)
reduce_mean_kernel(const float* __restrict__ row_loss, float* __restrict__ out)
{
    __shared__ float s[256];
    float acc = 0.0f;
    #pragma unroll
    for (int i = 0; i < TL_N / 256; ++i)
        acc += row_loss[threadIdx.x + 256 * i];
    s[threadIdx.x] = acc;
    __syncthreads();
    #pragma unroll
    for (int off = 128; off > 0; off >>= 1) {
        if (threadIdx.x < off) s[threadIdx.x] += s[threadIdx.x + off];
        __syncthreads();
    }
    if (threadIdx.x == 0)
        out[0] = s[0] * (1.0f / (float)TL_N);
}

extern "C" void kernel_launch(void* const* d_in, const int* in_sizes, int n_in,
                              void* d_out, int out_size, void* d_ws, size_t ws_size,
                              hipStream_t stream)
{
    const float* emb     = (const float*)d_in[0];
    // d_in[1] = labels (unused by the loss itself)
    const int*   pos_idx = (const int*)d_in[2];
    const int*   neg_idx = (const int*)d_in[3];
    float*       out     = (float*)d_out;
    float*       rowl    = (float*)d_ws;   // 4096 floats of scratch

    triplet_dist_kernel<<<dim3(TL_N / 16), dim3(256), 0, stream>>>(
        emb, pos_idx, neg_idx, rowl);
    reduce_mean_kernel<<<dim3(1), dim3(256), 0, stream>>>(rowl, out);
}